// SingleUnitSmallBlockBatch_28054726377748
// MI455X (gfx1250) — compile-verified
//
#include <hip/hip_runtime.h>
#include <math.h>

typedef __attribute__((ext_vector_type(2))) float v2f;
typedef __attribute__((ext_vector_type(8))) float v8f;

#define H 128
#define CIN 10

// ---------------------------------------------------------------- zero fill
__global__ void zero_u32_kernel(unsigned int* __restrict__ p, long long n) {
  long long i = (long long)blockIdx.x * blockDim.x + threadIdx.x;
  long long stride = (long long)gridDim.x * blockDim.x;
  for (; i < n; i += stride) p[i] = 0u;
}

// ------------------------------------------------------- input projection
// h[n,c] = relu(sum_k x[n,k] * Win[k,c] + bin[c]);  one block per node.
__global__ void input_proj_kernel(const float* __restrict__ x,
                                  const float* __restrict__ Win,
                                  const float* __restrict__ bin,
                                  float* __restrict__ h, int N) {
  int n = blockIdx.x;
  if (n >= N) return;
  int c = threadIdx.x;
  float acc = bin[c];
#pragma unroll
  for (int k = 0; k < CIN; ++k) acc += x[n * CIN + k] * Win[k * H + c];
  h[(long long)n * H + c] = fmaxf(acc, 0.f);
}

// ------------------------------------------------------------- scatter max
// values are >= 0 (post-relu), so u32 atomicMax on the bit pattern == float max
// with 0.0f init. 32 threads/edge, 4 channels each (float4 gather).
__global__ void scatter_max_kernel(const float* __restrict__ h,
                                   const int* __restrict__ src,
                                   const int* __restrict__ dst,
                                   unsigned int* __restrict__ agg, int E) {
  int idx = blockIdx.x * blockDim.x + threadIdx.x;
  int e = idx >> 5;
  if (e >= E) return;
  int q = idx & 31;
  int s = src[e], d = dst[e];
  float4 v = ((const float4*)h)[(long long)s * 32 + q];
  unsigned int* a = agg + (long long)d * H + q * 4;
  atomicMax(a + 0, __float_as_uint(v.x));
  atomicMax(a + 1, __float_as_uint(v.y));
  atomicMax(a + 2, __float_as_uint(v.z));
  atomicMax(a + 3, __float_as_uint(v.w));
}

// masked variant for the polygon subgraph; also marks sink nodes
__global__ void scatter_max_masked_kernel(const float* __restrict__ h,
                                          const int* __restrict__ src,
                                          const int* __restrict__ dst,
                                          const int* __restrict__ ea,
                                          const int* __restrict__ batch,
                                          unsigned int* __restrict__ agg,
                                          int* __restrict__ sink, int E) {
  int idx = blockIdx.x * blockDim.x + threadIdx.x;
  int e = idx >> 5;
  if (e >= E) return;
  if (ea[e] != 3) return;
  int s = src[e], d = dst[e];
  if (batch[s] != batch[d]) return;
  int q = idx & 31;
  float4 v = ((const float4*)h)[(long long)s * 32 + q];
  unsigned int* a = agg + (long long)d * H + q * 4;
  atomicMax(a + 0, __float_as_uint(v.x));
  atomicMax(a + 1, __float_as_uint(v.y));
  atomicMax(a + 2, __float_as_uint(v.z));
  atomicMax(a + 3, __float_as_uint(v.w));
  if (q == 0) sink[d] = 1;
}

// ------------------------------------------- WMMA GEMM + LayerNorm + fuse
// T = A @ W + bias ; y = LN(T)*g + beta ; out = relu(y + resid)
// if fused != null : fused += out * sigmoid(gate)
// Block: 256 threads = 8 waves; 16-row tile; wave w owns cols [16w,16w+16).
__global__ void gemm_ln_kernel(const float* __restrict__ A,
                               const float* __restrict__ W,
                               const float* __restrict__ bias,
                               const float* __restrict__ g,
                               const float* __restrict__ beta,
                               const float* resid,
                               float* out,
                               float* fused,
                               const float* gate,
                               int N) {
  __shared__ float sA[16][H + 4];
  __shared__ float sT[16][H + 4];
  const int row0 = blockIdx.x * 16;
  const int tid = threadIdx.x;

  // stage A tile into LDS (OOB rows -> 0)
  for (int i = tid; i < 16 * H; i += 256) {
    int r = i >> 7, c = i & (H - 1);
    int n = row0 + r;
    sA[r][c] = (n < N) ? A[(long long)n * H + c] : 0.f;
  }
  __syncthreads();

  const int wave = tid >> 5;
  const int lane = tid & 31;
  const int half = lane >> 4;   // 0: K pair (0,1), 1: K pair (2,3)
  const int l = lane & 15;      // M for A, N for B
  const int col0 = wave * 16;

  v8f acc = {};
#pragma unroll 8
  for (int k0 = 0; k0 < H; k0 += 4) {
    int ka = k0 + 2 * half;
    v2f a, b;
    a.x = sA[l][ka];
    a.y = sA[l][ka + 1];
    b.x = W[ka * H + col0 + l];
    b.y = W[(ka + 1) * H + col0 + l];
    acc = __builtin_amdgcn_wmma_f32_16x16x4_f32(
        /*neg_a=*/false, a, /*neg_b=*/false, b,
        /*c_mod=*/(short)0, acc, /*reuse_a=*/false, /*reuse_b=*/false);
  }

  // C/D layout: VGPR r -> M = r + 8*half, N = col0 + l
#pragma unroll
  for (int r = 0; r < 8; ++r)
    sT[r + 8 * half][col0 + l] = acc[r] + bias[col0 + l];
  __syncthreads();

  // LayerNorm: 16 threads per row, 8 channels each; rows map to half-waves,
  // so xor-shuffles up to 8 stay inside the row's 16-lane group.
  const int row = tid >> 4;
  const int sub = tid & 15;
  float v[8], s = 0.f, ss = 0.f;
#pragma unroll
  for (int j = 0; j < 8; ++j) {
    v[j] = sT[row][sub * 8 + j];
    s += v[j];
    ss += v[j] * v[j];
  }
#pragma unroll
  for (int m = 1; m < 16; m <<= 1) {
    s += __shfl_xor(s, m, 32);
    ss += __shfl_xor(ss, m, 32);
  }
  const float mu = s * (1.f / H);
  const float var = ss * (1.f / H) - mu * mu;
  const float rs = rsqrtf(var + 1e-5f);

  const int n = row0 + row;
  if (n < N) {
#pragma unroll
    for (int j = 0; j < 8; ++j) {
      int c = sub * 8 + j;
      float y = (v[j] - mu) * rs * g[c] + beta[c];
      float hn = fmaxf(y + resid[(long long)n * H + c], 0.f);
      out[(long long)n * H + c] = hn;
      if (fused) {
        float sg = 1.f / (1.f + __expf(-gate[c]));
        fused[(long long)n * H + c] += hn * sg;
      }
    }
  }
}

// ------------------------------------------------------------ pooling
__global__ void pool_kernel(const float* __restrict__ nf,
                            const float* __restrict__ subf,
                            const int* __restrict__ sink,
                            const float* __restrict__ x,
                            const int* __restrict__ batch,
                            unsigned int* __restrict__ v2b,
                            unsigned int* __restrict__ v3b,
                            int* __restrict__ flag2, int* __restrict__ flag3,
                            int N) {
  int idx = blockIdx.x * blockDim.x + threadIdx.x;
  int n = idx >> 5;
  if (n >= N) return;
  int q = idx & 31;
  int b = batch[n];
  if (sink[n]) {
    float4 v = ((const float4*)subf)[(long long)n * 32 + q];
    unsigned int* p = v2b + (long long)b * H + q * 4;
    atomicMax(p + 0, __float_as_uint(v.x));
    atomicMax(p + 1, __float_as_uint(v.y));
    atomicMax(p + 2, __float_as_uint(v.z));
    atomicMax(p + 3, __float_as_uint(v.w));
    if (q == 0) flag2[b] = 1;
  }
  if (x[(long long)n * CIN] <= 0.1f) {
    float4 v = ((const float4*)nf)[(long long)n * 32 + q];
    unsigned int* p = v3b + (long long)b * H + q * 4;
    atomicMax(p + 0, __float_as_uint(v.x));
    atomicMax(p + 1, __float_as_uint(v.y));
    atomicMax(p + 2, __float_as_uint(v.z));
    atomicMax(p + 3, __float_as_uint(v.w));
    if (q == 0) flag3[b] = 1;
  }
}

__global__ void finalize_kernel(const unsigned int* __restrict__ v2b,
                                const unsigned int* __restrict__ v3b,
                                const int* __restrict__ flag2,
                                const int* __restrict__ flag3,
                                float* __restrict__ out, int B) {
  int i = blockIdx.x * blockDim.x + threadIdx.x;
  if (i >= B * 2 * H) return;
  int b = i / (2 * H);
  int r = (i >> 7) & 1;
  int c = i & (H - 1);
  float val;
  if (r == 0)
    val = flag2[b] ? __uint_as_float(v2b[b * H + c]) : 1e-4f;
  else
    val = flag3[b] ? __uint_as_float(v3b[b * H + c]) : 1e-4f;
  out[i] = val;
}

// ---------------------------------------------------------------- launcher
extern "C" void kernel_launch(void* const* d_in, const int* in_sizes, int n_in,
                              void* d_out, int out_size, void* d_ws, size_t ws_size,
                              hipStream_t stream) {
  const float* x        = (const float*)d_in[0];
  const int*   eidx     = (const int*)d_in[1];
  const int*   eattr    = (const int*)d_in[2];
  const int*   batch    = (const int*)d_in[3];
  // d_in[4] = num_graphs scalar on device (B derived from out_size instead)
  const float* W_in     = (const float*)d_in[5];
  const float* b_in     = (const float*)d_in[6];
  const float* hop_W    = (const float*)d_in[7];
  const float* hop_b    = (const float*)d_in[8];
  const float* hop_g    = (const float*)d_in[9];
  const float* hop_beta = (const float*)d_in[10];
  const float* gates    = (const float*)d_in[11];
  const float* fin_W    = (const float*)d_in[12];
  const float* fin_b    = (const float*)d_in[13];
  const float* fin_g    = (const float*)d_in[14];
  const float* fin_beta = (const float*)d_in[15];

  const int N = in_sizes[0] / CIN;
  const int E = in_sizes[2];
  const int B = out_size / (2 * H);
  const int* src = eidx;
  const int* dst = eidx + E;

  const size_t nh = (size_t)N * H;
  float*        h     = (float*)d_ws;
  unsigned int* agg   = (unsigned int*)(h + nh);
  float*        fused = (float*)(agg + nh);
  float*        subf  = fused + nh;
  unsigned int* v2b   = (unsigned int*)(subf + nh);
  unsigned int* v3b   = v2b + (size_t)B * H;
  int*          sink  = (int*)(v3b + (size_t)B * H);
  int*          flag2 = sink + N;
  int*          flag3 = flag2 + B;

  const int ZT = 256;
  // zero fused (nh) and the contiguous tail: v2b, v3b, sink, flag2, flag3
  long long tail = (long long)2 * B * H + N + 2 * B;
  zero_u32_kernel<<<(unsigned)((nh + ZT - 1) / ZT), ZT, 0, stream>>>((unsigned int*)fused, (long long)nh);
  zero_u32_kernel<<<(unsigned)((tail + ZT - 1) / ZT), ZT, 0, stream>>>(v2b, tail);

  // input projection
  input_proj_kernel<<<N, H, 0, stream>>>(x, W_in, b_in, h, N);

  const unsigned gScat = (unsigned)(((long long)E * 32 + 255) / 256);
  const unsigned gGemm = (unsigned)((N + 15) / 16);

  // three message-passing hops
  for (int i = 0; i < 3; ++i) {
    zero_u32_kernel<<<(unsigned)((nh + ZT - 1) / ZT), ZT, 0, stream>>>(agg, (long long)nh);
    scatter_max_kernel<<<gScat, 256, 0, stream>>>(h, src, dst, agg, E);
    gemm_ln_kernel<<<gGemm, 256, 0, stream>>>(
        (const float*)agg, hop_W + (size_t)i * H * H, hop_b + i * H,
        hop_g + i * H, hop_beta + i * H,
        /*resid=*/h, /*out=*/h, /*fused=*/fused, /*gate=*/gates + i * H, N);
  }

  // polygon-subgraph final hop
  zero_u32_kernel<<<(unsigned)((nh + ZT - 1) / ZT), ZT, 0, stream>>>(agg, (long long)nh);
  scatter_max_masked_kernel<<<gScat, 256, 0, stream>>>(fused, src, dst, eattr,
                                                       batch, agg, sink, E);
  gemm_ln_kernel<<<gGemm, 256, 0, stream>>>(
      (const float*)agg, fin_W, fin_b, fin_g, fin_beta,
      /*resid=*/fused, /*out=*/subf, /*fused=*/nullptr, /*gate=*/nullptr, N);

  // pooling + finalize
  const unsigned gPool = (unsigned)(((long long)N * 32 + 255) / 256);
  pool_kernel<<<gPool, 256, 0, stream>>>(fused, subf, sink, x, batch,
                                         v2b, v3b, flag2, flag3, N);
  finalize_kernel<<<(unsigned)((B * 2 * H + 255) / 256), 256, 0, stream>>>(
      v2b, v3b, flag2, flag3, (float*)d_out, B);
}